// AffineTransform2D_2748779070090
// MI455X (gfx1250) — compile-verified
//
#include <hip/hip_runtime.h>

// Spatial-transformer affine bilinear sampling, MI455X (gfx1250, wave32).
//
// im:     [mb, 512, 512, 1, 3] f32
// thetas: [mb, 6]              f32
// out:    [mb, 512, 512, 3]    f32
//
// Memory-bound gather (~200 MB DRAM traffic -> ~9us at 23.3 TB/s).
// One pixel per lane; thetas are wave-uniform (scalar loads); taps merge
// into global_load_b96; output uses non-temporal stores so the streamed
// 100 MB result does not evict the (L2-resident, 100 MB < 192 MB) input.

#define H_IMG 512
#define W_IMG 512

__global__ __launch_bounds__(256) void affine_bilinear_gfx1250(
    const float* __restrict__ im,
    const float* __restrict__ th,
    float* __restrict__ out)
{
    const int b = blockIdx.y;                       // image index (uniform)
    const int p = blockIdx.x * 256 + threadIdx.x;   // pixel index within image
    const int px = p & (W_IMG - 1);
    const int py = p >> 9;                          // log2(W_IMG)

    // Wave-uniform theta coefficients -> compiler emits s_load (SMEM/KMcnt path)
    const float* tb = th + b * 6;
    const float t0 = tb[0], t1 = tb[1], t2 = tb[2];
    const float t3 = tb[3], t4 = tb[4], t5 = tb[5];

    // normalized grid coords: linspace(-1, 1, N) -> -1 + i * 2/(N-1)
    const float Xg = fmaf((float)px, 2.0f / (float)(W_IMG - 1), -1.0f);
    const float Yg = fmaf((float)py, 2.0f / (float)(H_IMG - 1), -1.0f);

    // affine transform (the whole "einsum": 4 FMAs per lane)
    float X = fmaf(t0, Xg, fmaf(t1, Yg, t2));
    float Y = fmaf(t3, Xg, fmaf(t4, Yg, t5));

    // to pixel space: (v + 1) / 2 * N
    X = (X + 1.0f) * (0.5f * (float)W_IMG);
    Y = (Y + 1.0f) * (0.5f * (float)H_IMG);

    int x0 = (int)floorf(X);
    int y0 = (int)floorf(Y);
    int x1 = x0 + 1;
    int y1 = y0 + 1;
    x0 = min(max(x0, 0), W_IMG - 1);
    x1 = min(max(x1, 0), W_IMG - 1);
    y0 = min(max(y0, 0), H_IMG - 1);
    y1 = min(max(y1, 0), H_IMG - 1);

    // bilinear weights from CLIPPED integer coords (matches reference exactly)
    const float x0f = (float)x0, x1f = (float)x1;
    const float y0f = (float)y0, y1f = (float)y1;
    const float wa = (x1f - X) * (y1f - Y);
    const float wb = (x1f - X) * (Y - y0f);
    const float wc = (X - x0f) * (y1f - Y);
    const float wd = (X - x0f) * (Y - y0f);

    const int base = b * (H_IMG * W_IMG);

    // 4 taps x 3 contiguous floats; backend merges each into global_load_b96
    const float* pa = im + 3 * (base + y0 * W_IMG + x0);
    const float* pb = im + 3 * (base + y1 * W_IMG + x0);
    const float* pc = im + 3 * (base + y0 * W_IMG + x1);
    const float* pd = im + 3 * (base + y1 * W_IMG + x1);

    const float a0 = pa[0], a1 = pa[1], a2 = pa[2];
    const float b0 = pb[0], b1 = pb[1], b2 = pb[2];
    const float c0 = pc[0], c1 = pc[1], c2 = pc[2];
    const float d0 = pd[0], d1 = pd[1], d2 = pd[2];

    float o0 = wa * a0;
    float o1 = wa * a1;
    float o2 = wa * a2;
    o0 = fmaf(wb, b0, o0);  o1 = fmaf(wb, b1, o1);  o2 = fmaf(wb, b2, o2);
    o0 = fmaf(wc, c0, o0);  o1 = fmaf(wc, c1, o1);  o2 = fmaf(wc, c2, o2);
    o0 = fmaf(wd, d0, o0);  o1 = fmaf(wd, d1, o1);  o2 = fmaf(wd, d2, o2);

    // Streaming write-once output: non-temporal (TH=NT) so it does not
    // evict the L2-resident input image. Consecutive lanes -> contiguous
    // 384 B per wave.
    float* po = out + 3 * (base + p);
    __builtin_nontemporal_store(o0, po + 0);
    __builtin_nontemporal_store(o1, po + 1);
    __builtin_nontemporal_store(o2, po + 2);
}

extern "C" void kernel_launch(void* const* d_in, const int* in_sizes, int n_in,
                              void* d_out, int out_size, void* d_ws, size_t ws_size,
                              hipStream_t stream) {
    const float* im     = (const float*)d_in[0];   // [mb,512,512,1,3] f32
    // d_in[1] = mb_size scalar (host-known via in_sizes), unused on device
    const float* thetas = (const float*)d_in[2];   // [mb,6] f32
    float* out          = (float*)d_out;           // [mb,512,512,3] f32

    const int mb = in_sizes[2] / 6;                // thetas flat count / 6

    dim3 block(256);                               // 8 wave32 waves
    dim3 grid((H_IMG * W_IMG) / 256, mb);          // 1024 x mb blocks
    affine_bilinear_gfx1250<<<grid, block, 0, stream>>>(im, thetas, out);
}